// GCN_8177617732163
// MI455X (gfx1250) — compile-verified
//
#include <hip/hip_runtime.h>
#include <hip/hip_bf16.h>

typedef __attribute__((ext_vector_type(2))) float v2f;
typedef __attribute__((ext_vector_type(8))) float v8f;

#define N_NODES  100000
#define N_EDGES  1250000
#define CH       64
#define N_GRAPHS 512
#define OUT_CH   8

// ---------------- utility ----------------
__global__ void gcn_zero_f32(float* __restrict__ p, int n) {
    int i = blockIdx.x * blockDim.x + threadIdx.x;
    if (i < n) p[i] = 0.0f;
}

// deg[dst] += 1 per edge (self-loop +1 folded into dinv kernel)
__global__ void gcn_deg(const int* __restrict__ dst, float* __restrict__ deg) {
    int e = blockIdx.x * blockDim.x + threadIdx.x;
    if (e < N_EDGES) atomicAdd(&deg[dst[e]], 1.0f);
}

__global__ void gcn_dinv(float* __restrict__ deg_dinv) {
    int n = blockIdx.x * blockDim.x + threadIdx.x;
    if (n < N_NODES) deg_dinv[n] = rsqrtf(deg_dinv[n] + 1.0f);
}

// ---------------- H = X @ W  (fp32 WMMA 16x16x4) ----------------
// Block = 256 threads = 8 waves; each wave computes a 16x64 output strip.
// W staged in LDS pre-paired along K: Wp[k/2][n] = {W[k][n], W[k+1][n]}
// so each B fragment is one aligned ds_load_b64 into an even VGPR pair.
__global__ void __launch_bounds__(256) gcn_gemm64(const float* __restrict__ X,
                                                  const float* __restrict__ W,
                                                  float* __restrict__ H) {
    __shared__ float2 Wp[(CH / 2) * CH];    // 16 KB
    for (int i = threadIdx.x; i < CH * CH; i += 256) {
        const int k = i >> 6, n = i & 63;
        // float2 element ((k>>1)*CH + n), component (k&1)
        ((float*)&Wp[(k >> 1) * CH + n])[k & 1] = W[i];
    }
    __syncthreads();

    const int wave = threadIdx.x >> 5;
    const int lane = threadIdx.x & 31;
    const int half = lane >> 4;             // 0: K0/K1 ; 1: K2/K3
    const int l16  = lane & 15;
    const int row0 = (blockIdx.x * 8 + wave) * 16;

    // A-fragment source row for this lane (clamped; padded rows' stores are guarded)
    int arow = row0 + l16;
    if (arow >= N_NODES) arow = N_NODES - 1;
    const float* __restrict__ xrow = X + (size_t)arow * CH;

    v8f acc0 = {}, acc1 = {}, acc2 = {}, acc3 = {};
    #pragma unroll
    for (int kb = 0; kb < CH; kb += 4) {
        const int ka = kb + half * 2;                     // even
        const v2f a = *(const v2f*)(xrow + ka);           // global_load_b64
        const v2f* __restrict__ wb = (const v2f*)&Wp[(ka >> 1) * CH];
        const v2f b0 = wb[      l16];                     // ds_load_b64 each
        const v2f b1 = wb[16 +  l16];
        const v2f b2 = wb[32 +  l16];
        const v2f b3 = wb[48 +  l16];
        acc0 = __builtin_amdgcn_wmma_f32_16x16x4_f32(false, a, false, b0, (short)0, acc0, false, false);
        acc1 = __builtin_amdgcn_wmma_f32_16x16x4_f32(false, a, false, b1, (short)0, acc1, false, false);
        acc2 = __builtin_amdgcn_wmma_f32_16x16x4_f32(false, a, false, b2, (short)0, acc2, false, false);
        acc3 = __builtin_amdgcn_wmma_f32_16x16x4_f32(false, a, false, b3, (short)0, acc3, false, false);
    }

    // C/D layout: lanes 0-15 -> rows row0+v, lanes 16-31 -> rows row0+8+v; col = tile*16 + l16
    const int rbase = row0 + half * 8;
    #pragma unroll
    for (int v = 0; v < 8; ++v) {
        const int r = rbase + v;
        if (r < N_NODES) {
            float* o = H + (size_t)r * CH + l16;
            o[0]  = acc0[v];
            o[16] = acc1[v];
            o[32] = acc2[v];
            o[48] = acc3[v];
        }
    }
}

// ---------------- edge scatter: agg[dst] += coef * H[src] ----------------
// 16 threads per edge, float4 per thread -> coalesced gather, f32 atomics scatter.
__global__ void gcn_scatter(const int* __restrict__ src, const int* __restrict__ dst,
                            const float* __restrict__ dinv, const float* __restrict__ H,
                            float* __restrict__ agg) {
    long long idx = (long long)blockIdx.x * blockDim.x + threadIdx.x;
    if (idx >= (long long)N_EDGES * 16) return;
    const int e  = (int)(idx >> 4);
    const int c4 = (int)(idx & 15) * 4;
    const int s = src[e], d = dst[e];
    const float coef = dinv[s] * dinv[d];
    const float4 h = *(const float4*)(H + (size_t)s * CH + c4);
    float* ap = agg + (size_t)d * CH + c4;
    atomicAdd(ap + 0, h.x * coef);
    atomicAdd(ap + 1, h.y * coef);
    atomicAdd(ap + 2, h.z * coef);
    atomicAdd(ap + 3, h.w * coef);
}

// ---------------- out = relu(agg + H*dinv^2 + b)  (in-place into agg OK) ----
__global__ void gcn_finalize(const float* __restrict__ H, const float* __restrict__ agg,
                             const float* __restrict__ dinv, const float* __restrict__ b,
                             float* __restrict__ out) {
    int i = blockIdx.x * blockDim.x + threadIdx.x;
    if (i >= N_NODES * CH) return;
    const int n = i >> 6, c = i & 63;
    const float di = dinv[n];
    const float v = agg[i] + H[i] * di * di + b[c];
    out[i] = v > 0.0f ? v : 0.0f;
}

// ---------------- pooling ----------------
__global__ void gcn_pool_sum(const float* __restrict__ F, const int* __restrict__ batch,
                             float* __restrict__ sums) {
    int i = blockIdx.x * blockDim.x + threadIdx.x;
    if (i >= N_NODES * CH) return;
    const int n = i >> 6, c = i & 63;
    atomicAdd(&sums[batch[n] * CH + c], F[i]);
}

__global__ void gcn_pool_cnt(const int* __restrict__ batch, float* __restrict__ cnt) {
    int n = blockIdx.x * blockDim.x + threadIdx.x;
    if (n < N_NODES) atomicAdd(&cnt[batch[n]], 1.0f);
}

// ---------------- sigmoid(pooled @ Wfc + bfc) ----------------
__global__ void gcn_fc(const float* __restrict__ sums, const float* __restrict__ cnt,
                       const float* __restrict__ Wfc, const float* __restrict__ bfc,
                       float* __restrict__ out) {
    int i = blockIdx.x * blockDim.x + threadIdx.x;
    if (i >= N_GRAPHS * OUT_CH) return;
    const int g = i >> 3, o = i & 7;
    float c = cnt[g]; c = c > 1.0f ? c : 1.0f;
    const float inv = 1.0f / c;
    float acc = bfc[o];
    #pragma unroll 8
    for (int k = 0; k < CH; ++k) acc += sums[g * CH + k] * inv * Wfc[k * OUT_CH + o];
    out[i] = 1.0f / (1.0f + expf(-acc));
}

extern "C" void kernel_launch(void* const* d_in, const int* in_sizes, int n_in,
                              void* d_out, int out_size, void* d_ws, size_t ws_size,
                              hipStream_t stream) {
    const float* x    = (const float*)d_in[0];
    const int*   edge = (const int*)d_in[1];
    const int*   src  = edge;
    const int*   dst  = edge + N_EDGES;
    const int*   batch = (const int*)d_in[2];
    const float* W0 = (const float*)d_in[3];
    const float* b0 = (const float*)d_in[4];
    const float* W1 = (const float*)d_in[5];
    const float* b1 = (const float*)d_in[6];
    const float* W2 = (const float*)d_in[7];
    const float* b2 = (const float*)d_in[8];
    const float* Wfc = (const float*)d_in[9];
    const float* bfc = (const float*)d_in[10];

    // workspace layout (floats)
    const size_t NC = (size_t)N_NODES * CH;       // 6.4M floats
    float* ws   = (float*)d_ws;
    float* dinv = ws;                              // N_NODES
    float* H    = ws + 100352;                     // NC (aligned start)
    float* P    = H + NC;                          // NC
    float* Q    = P + NC;                          // NC
    float* sums = Q + NC;                          // 512*64
    float* cnt  = sums + N_GRAPHS * CH;            // 512

    const int T = 256;
    const int gN   = (N_NODES + T - 1) / T;
    const int gE   = (N_EDGES + T - 1) / T;
    const int gNC  = (int)((NC + T - 1) / T);
    const int gE16 = (int)(((long long)N_EDGES * 16 + T - 1) / T);
    const int gGemm = (N_NODES + 127) / 128;       // 128 rows per block (8 waves x 16)

    // degrees -> dinv
    gcn_zero_f32<<<gN, T, 0, stream>>>(dinv, N_NODES);
    gcn_deg<<<gE, T, 0, stream>>>(dst, dinv);
    gcn_dinv<<<gN, T, 0, stream>>>(dinv);

    // layer 0: x -> P
    gcn_gemm64<<<gGemm, T, 0, stream>>>(x, W0, H);
    gcn_zero_f32<<<gNC, T, 0, stream>>>(P, (int)NC);
    gcn_scatter<<<gE16, T, 0, stream>>>(src, dst, dinv, H, P);
    gcn_finalize<<<gNC, T, 0, stream>>>(H, P, dinv, b0, P);

    // layer 1: P -> Q
    gcn_gemm64<<<gGemm, T, 0, stream>>>(P, W1, H);
    gcn_zero_f32<<<gNC, T, 0, stream>>>(Q, (int)NC);
    gcn_scatter<<<gE16, T, 0, stream>>>(src, dst, dinv, H, Q);
    gcn_finalize<<<gNC, T, 0, stream>>>(H, Q, dinv, b1, Q);

    // layer 2: Q -> P
    gcn_gemm64<<<gGemm, T, 0, stream>>>(Q, W2, H);
    gcn_zero_f32<<<gNC, T, 0, stream>>>(P, (int)NC);
    gcn_scatter<<<gE16, T, 0, stream>>>(src, dst, dinv, H, P);
    gcn_finalize<<<gNC, T, 0, stream>>>(H, P, dinv, b2, P);

    // pooling + FC + sigmoid
    gcn_zero_f32<<<(N_GRAPHS * CH + N_GRAPHS + T - 1) / T, T, 0, stream>>>(sums, N_GRAPHS * CH + N_GRAPHS);
    gcn_pool_sum<<<gNC, T, 0, stream>>>(P, batch, sums);
    gcn_pool_cnt<<<gN, T, 0, stream>>>(batch, cnt);
    gcn_fc<<<(N_GRAPHS * OUT_CH + T - 1) / T, T, 0, stream>>>(sums, cnt, Wfc, bfc, (float*)d_out);
}